// Attention5_70841190580884
// MI455X (gfx1250) — compile-verified
//
#include <hip/hip_runtime.h>
#include <math.h>

typedef __attribute__((ext_vector_type(2))) float v2f;
typedef __attribute__((ext_vector_type(8))) float v8f;

#define WMMA_F32(a, b, c) \
  __builtin_amdgcn_wmma_f32_16x16x4_f32(false, (a), false, (b), (short)0, (c), false, false)

constexpr int B_ = 32, S_ = 2048, H_ = 256, Z_ = 64;
constexpr int SPLIT_ = 16;
constexpr float TEMP_INV_ = 1.0f / 16.0f;   // 1/sqrt(256)
constexpr float NEG_INF_ = -1e9f;

// ---------------------------------------------------------------------------
// Projection GEMM: C[M,N] = A[M,K] @ W[K,N], fp32 WMMA 16x16x4.
// Workgroup = 256 threads (8 waves), computes a 128 x (NT*16) block.
// ---------------------------------------------------------------------------
template <int K, int NT>
__global__ void proj_gemm(const float* __restrict__ A, const float* __restrict__ W,
                          float* __restrict__ C, int M, int N) {
  constexpr int AST = K + 4;
  constexpr int WN  = NT * 16;
  constexpr int WST = WN + 8;
  extern __shared__ float lds[];
  float* As = lds;                 // [128][AST]
  float* Ws = lds + 128 * AST;     // [K][WST]

  const int tid  = threadIdx.x;
  const int lane = tid & 31;
  const int w    = tid >> 5;       // wave id 0..7 -> i-tile
  const int m    = lane & 15;
  const int hi   = lane >> 4;      // lane half
  const int kh   = hi * 2;

  const long long i0 = (long long)blockIdx.x * 128;
  const int n0 = blockIdx.y * WN;

  for (int idx = tid; idx < 128 * K; idx += 256) {
    int r = idx / K, c = idx % K;
    As[r * AST + c] = A[(i0 + r) * K + c];
  }
  for (int idx = tid; idx < K * WN; idx += 256) {
    int r = idx / WN, c = idx % WN;
    Ws[r * WST + c] = W[(long long)r * N + n0 + c];
  }
  __syncthreads();

  v8f zero = {};
  v8f acc[NT];
#pragma unroll
  for (int jx = 0; jx < NT; ++jx) acc[jx] = zero;

  const float* arow = &As[(w * 16 + m) * AST];
  for (int k0 = 0; k0 < K; k0 += 4) {
    v2f a = *(const v2f*)(arow + k0 + kh);           // A frag: (row m, k0+kh .. +1)
#pragma unroll
    for (int jx = 0; jx < NT; ++jx) {
      v2f bf;
      bf.x = Ws[(k0 + kh)     * WST + jx * 16 + m];  // B frag: W[k][n]
      bf.y = Ws[(k0 + kh + 1) * WST + jx * 16 + m];
      acc[jx] = WMMA_F32(a, bf, acc[jx]);
    }
  }

#pragma unroll
  for (int jx = 0; jx < NT; ++jx) {
#pragma unroll
    for (int r = 0; r < 8; ++r) {
      long long row = i0 + w * 16 + r + hi * 8;      // C/D layout: M = r + 8*half
      C[row * N + n0 + jx * 16 + m] = acc[jx][r];
    }
  }
}

// ---------------------------------------------------------------------------
// scores[b,i,j] = mask[b,i] ? (q_i . k_j)/16 : -1e9, into d_out's attn region.
// 128x128 tile per workgroup. Two pipeline levels:
//  - chunk level: 8 K-chunks of 32, double-buffered LDS (72KB), next chunk
//    prefetched from global into registers during the 64-WMMA compute phase;
//  - k-step level: A-frag + 8 B-frags for step ks+1 preloaded from LDS into a
//    second register set before the 8 WMMAs of step ks, so ds waits land after
//    a full WMMA group instead of before each one.
// ---------------------------------------------------------------------------
__global__ void qk_scores(const float* __restrict__ Q, const float* __restrict__ Kmat,
                          const int* __restrict__ mask, float* __restrict__ scores) {
  constexpr int KC  = 32;            // k-chunk width
  constexpr int NC  = H_ / KC;       // 8 chunks
  constexpr int NKS = KC / 4;        // 8 k-steps per chunk
  constexpr int AST = KC + 4;        // 36: conflict-free lane-row float2 reads
  constexpr int BUF = 128 * AST;     // floats per tile buffer
  extern __shared__ float lds[];
  float* Qs = lds;                   // [2][128][36]
  float* Ks = lds + 2 * BUF;         // [2][128][36]

  const int tid  = threadIdx.x;
  const int lane = tid & 31;
  const int w    = tid >> 5;
  const int m    = lane & 15;
  const int hi   = lane >> 4;
  const int kh   = hi * 2;

  const int b = blockIdx.z;
  const long long i0 = (long long)blockIdx.x * 128;
  const long long j0 = (long long)blockIdx.y * 128;
  const float* Qb = Q    + (long long)b * S_ * H_;
  const float* Kb = Kmat + (long long)b * S_ * H_;

  // Staging map: 1024 float4 per (matrix, chunk); thread handles fid = tid+256*t.
  const int rbase = tid >> 3;
  const int ccol  = (tid & 7) * 4;

  v8f zero = {};
  v8f acc[8];
#pragma unroll
  for (int jx = 0; jx < 8; ++jx) acc[jx] = zero;

  // ---- prologue: stage chunk 0 into buffer 0
  {
    float4 qr[4], kr[4];
#pragma unroll
    for (int t = 0; t < 4; ++t) {
      const long long r = rbase + 32 * t;
      qr[t] = *(const float4*)(Qb + (i0 + r) * H_ + ccol);
      kr[t] = *(const float4*)(Kb + (j0 + r) * H_ + ccol);
    }
#pragma unroll
    for (int t = 0; t < 4; ++t) {
      const int r = rbase + 32 * t;
      *(float4*)(Qs + r * AST + ccol) = qr[t];
      *(float4*)(Ks + r * AST + ccol) = kr[t];
    }
  }
  __syncthreads();

  for (int ch = 0; ch < NC; ++ch) {
    const int buf = ch & 1;
    float4 qr[4], kr[4];
    if (ch + 1 < NC) {                       // global prefetch of next chunk
      const int koff = (ch + 1) * KC + ccol;
#pragma unroll
      for (int t = 0; t < 4; ++t) {
        const long long r = rbase + 32 * t;
        qr[t] = *(const float4*)(Qb + (i0 + r) * H_ + koff);
        kr[t] = *(const float4*)(Kb + (j0 + r) * H_ + koff);
      }
    }

    const float* qrow = Qs + buf * BUF + (w * 16 + m) * AST + kh;
    const float* krow = Ks + buf * BUF + m * AST + kh;

    // k-step software pipeline: fragment registers double-buffered
    v2f afr[2];
    v2f bfr[2][8];
    afr[0] = *(const v2f*)(qrow);
#pragma unroll
    for (int jx = 0; jx < 8; ++jx)
      bfr[0][jx] = *(const v2f*)(krow + jx * 16 * AST);

#pragma unroll
    for (int ks = 0; ks < NKS; ++ks) {
      const int cur = ks & 1;
      const int nxt = cur ^ 1;
      if (ks + 1 < NKS) {                    // preload next k-step fragments
        const int koff = (ks + 1) * 4;
        afr[nxt] = *(const v2f*)(qrow + koff);
#pragma unroll
        for (int jx = 0; jx < 8; ++jx)
          bfr[nxt][jx] = *(const v2f*)(krow + jx * 16 * AST + koff);
      }
#pragma unroll
      for (int jx = 0; jx < 8; ++jx)
        acc[jx] = WMMA_F32(afr[cur], bfr[cur][jx], acc[jx]);
    }

    if (ch + 1 < NC) {
      __syncthreads();                       // all readers of the other buffer done
      const int nb = (ch + 1) & 1;
#pragma unroll
      for (int t = 0; t < 4; ++t) {
        const int r = rbase + 32 * t;
        *(float4*)(Qs + nb * BUF + r * AST + ccol) = qr[t];
        *(float4*)(Ks + nb * BUF + r * AST + ccol) = kr[t];
      }
      __syncthreads();                       // next buffer ready
    }
  }

  const int* mb = mask + b * S_;
  float* sb = scores + (long long)b * S_ * S_;
#pragma unroll
  for (int r = 0; r < 8; ++r) {
    long long i = i0 + w * 16 + r + hi * 8;
    bool on = (mb[i] != 0);
#pragma unroll
    for (int jx = 0; jx < 8; ++jx) {
      float vsc = on ? acc[jx][r] * TEMP_INV_ : NEG_INF_;
      sb[i * S_ + j0 + jx * 16 + m] = vsc;
    }
  }
}

// ---------------------------------------------------------------------------
// Softmax over the QUERY axis (i): per-(b,j) column online max/sum, split over
// 16 row-chunks for parallelism. Coalesced row reads.
// ---------------------------------------------------------------------------
__global__ void col_stats_partial(const float* __restrict__ scores,
                                  float* __restrict__ pmax, float* __restrict__ psum) {
  const int b = blockIdx.x;
  const int j = blockIdx.y * 256 + threadIdx.x;
  const int split = blockIdx.z;
  constexpr int ROWS = S_ / SPLIT_;   // 128
  const float* sb = scores + ((long long)b * S_ + (long long)split * ROWS) * S_ + j;
  float m = -3.4e38f, sum = 0.0f;
  for (int r = 0; r < ROWS; ++r) {
    if (r + 8 < ROWS) __builtin_prefetch(sb + (long long)(r + 8) * S_, 0, 1);
    float s = sb[(long long)r * S_];
    float mn = fmaxf(m, s);
    sum = sum * __expf(m - mn) + __expf(s - mn);
    m = mn;
  }
  long long o = ((long long)b * S_ + j) * SPLIT_ + split;
  pmax[o] = m;
  psum[o] = sum;
}

__global__ void col_stats_combine(const float* __restrict__ pmax,
                                  const float* __restrict__ psum,
                                  float* __restrict__ cmax, float* __restrict__ cinv) {
  long long idx = (long long)blockIdx.x * 256 + threadIdx.x;  // = b*S + j
  const float* pm = pmax + idx * SPLIT_;
  const float* ps = psum + idx * SPLIT_;
  float m = -3.4e38f;
#pragma unroll
  for (int p = 0; p < SPLIT_; ++p) m = fmaxf(m, pm[p]);
  float sum = 0.0f;
#pragma unroll
  for (int p = 0; p < SPLIT_; ++p) sum += ps[p] * __expf(pm[p] - m);
  cmax[idx] = m;
  cinv[idx] = 1.0f / sum;
}

// ---------------------------------------------------------------------------
// Fused: attn = exp(score - colmax[j]) * cinv[j] (in-place in d_out's attn
// region) and output = attn @ V via WMMA. Wave owns a 16-row i-tile and all
// 64 z-columns (4 accumulators), streaming j in steps of 4.
// ---------------------------------------------------------------------------
__global__ void attn_av(float* __restrict__ scores,           // raw in -> attn out
                        const float* __restrict__ V,
                        const float* __restrict__ cmax, const float* __restrict__ cinv,
                        float* __restrict__ out) {
  const int tid  = threadIdx.x;
  const int lane = tid & 31;
  const int w    = tid >> 5;
  const int m    = lane & 15;
  const int hi   = lane >> 4;
  const int kh   = hi * 2;

  const int b = blockIdx.y;
  const long long i0 = (long long)(blockIdx.x * 8 + w) * 16;
  float* sb       = scores + (long long)b * S_ * S_;
  const float* vb = V      + (long long)b * S_ * Z_;
  const float* cm = cmax   + (long long)b * S_;
  const float* ci = cinv   + (long long)b * S_;

  v8f zero = {};
  v8f acc[4];
#pragma unroll
  for (int zx = 0; zx < 4; ++zx) acc[zx] = zero;

  float* srow = sb + (i0 + m) * S_;
  for (int j0 = 0; j0 < S_; j0 += 4) {
    const int j = j0 + kh;
    v2f sc = *(const v2f*)(srow + j);
    v2f a;
    a.x = __expf(sc.x - cm[j])     * ci[j];
    a.y = __expf(sc.y - cm[j + 1]) * ci[j + 1];
    *(v2f*)(srow + j) = a;                          // materialize normalized attn
#pragma unroll
    for (int zx = 0; zx < 4; ++zx) {
      v2f bf;
      bf.x = vb[(long long)j       * Z_ + zx * 16 + m];
      bf.y = vb[(long long)(j + 1) * Z_ + zx * 16 + m];
      acc[zx] = WMMA_F32(a, bf, acc[zx]);
    }
  }

#pragma unroll
  for (int zx = 0; zx < 4; ++zx) {
#pragma unroll
    for (int r = 0; r < 8; ++r) {
      long long row = i0 + r + hi * 8;
      out[((long long)b * S_ + row) * Z_ + zx * 16 + m] = acc[zx][r];
    }
  }
}

// ---------------------------------------------------------------------------
extern "C" void kernel_launch(void* const* d_in, const int* in_sizes, int n_in,
                              void* d_out, int out_size, void* d_ws, size_t ws_size,
                              hipStream_t stream) {
  (void)in_sizes; (void)n_in; (void)out_size; (void)ws_size;

  const float* enc  = (const float*)d_in[0];   // encoder_hidden_states [B,S,H]
  const float* dec  = (const float*)d_in[1];   // decoder_hidden_states [B,S,H]
  /* d_in[2] decoder_final_hidden_state: unused by reference */
  const float* lat  = (const float*)d_in[3];   // latent_z_seq [B,S,Z]
  const int*   mask = (const int*)d_in[4];     // mask [B,S]
  const float* Wq   = (const float*)d_in[5];
  const float* Wk   = (const float*)d_in[6];
  const float* Wv   = (const float*)d_in[7];

  float* out  = (float*)d_out;                          // [B,S,Z]
  float* attn = out + (long long)B_ * S_ * Z_;          // [B,S,S] scores -> attn

  float* ws = (float*)d_ws;
  long long o = 0;
  float* q    = ws + o; o += (long long)B_ * S_ * H_;
  float* k    = ws + o; o += (long long)B_ * S_ * H_;
  float* v    = ws + o; o += (long long)B_ * S_ * Z_;
  float* pmax = ws + o; o += (long long)B_ * S_ * SPLIT_;
  float* psum = ws + o; o += (long long)B_ * S_ * SPLIT_;
  float* cmax = ws + o; o += (long long)B_ * S_;
  float* cinv = ws + o;

  const int M = B_ * S_;

  size_t ldsPH = (size_t)(128 * (H_ + 4) + H_ * (8 * 16 + 8)) * sizeof(float);
  size_t ldsPZ = (size_t)(128 * (Z_ + 4) + Z_ * (4 * 16 + 8)) * sizeof(float);
  size_t ldsQK = (size_t)(4 * 128 * (32 + 4)) * sizeof(float);   // 73,728 B

  hipFuncSetAttribute(reinterpret_cast<const void*>(&proj_gemm<256, 8>),
                      hipFuncAttributeMaxDynamicSharedMemorySize, (int)ldsPH);
  hipFuncSetAttribute(reinterpret_cast<const void*>(&proj_gemm<64, 4>),
                      hipFuncAttributeMaxDynamicSharedMemorySize, (int)ldsPZ);
  hipFuncSetAttribute(reinterpret_cast<const void*>(&qk_scores),
                      hipFuncAttributeMaxDynamicSharedMemorySize, (int)ldsQK);

  // Projections
  proj_gemm<256, 8><<<dim3(M / 128, H_ / 128), 256, ldsPH, stream>>>(dec, Wq, q, M, H_);
  proj_gemm<256, 8><<<dim3(M / 128, H_ / 128), 256, ldsPH, stream>>>(enc, Wk, k, M, H_);
  proj_gemm<64, 4><<<dim3(M / 128, 1), 256, ldsPZ, stream>>>(lat, Wv, v, M, Z_);

  // Masked scaled scores -> d_out attn region (double-buffered WMMA GEMM)
  qk_scores<<<dim3(S_ / 128, S_ / 128, B_), 256, ldsQK, stream>>>(q, k, mask, attn);

  // Column-wise (query-axis) softmax statistics
  col_stats_partial<<<dim3(B_, S_ / 256, SPLIT_), 256, 0, stream>>>(attn, pmax, psum);
  col_stats_combine<<<dim3((B_ * S_) / 256), 256, 0, stream>>>(pmax, psum, cmax, cinv);

  // Fused normalize (attn in-place) + attn @ V
  attn_av<<<dim3(S_ / 128, B_), 256, 0, stream>>>(attn, v, cmax, cinv, out);
}